// SparseMemory_52441550684833
// MI455X (gfx1250) — compile-verified
//
#include <hip/hip_runtime.h>
#include <hip/hip_bf16.h>
#include <math.h>

// Problem constants (from reference)
#define BATCH   32
#define MMEM    100000
#define WCELL   64
#define RREADS  9          // K_READS + 1
#define IFACE_N 138        // 2*WCELL + R + 1
#define DIN     512

// ws layout (float units)
#define WS_IFACE   0                    // 32*138 = 4416, padded
#define WS_NEWROW  4608                 // 32*9*64 = 18432
#define WS_SCORES  (4608 + 18432)       // 32*100000 = 3,200,000

typedef float v8f __attribute__((ext_vector_type(8)));
typedef float v2f __attribute__((ext_vector_type(2)));

// ---------------------------------------------------------------------------
// Kernel A: iface = xi(32x512) @ W(512x138) + b_lin, via V_WMMA_F32_16X16X4_F32
// 1 block, 256 threads = 8 waves. 18 (mt,nt) tiles of 16x16, K looped in 4s.
// A-frag (16x4 f32): lane l<16 -> M=l, v0:K=0 v1:K=1; lane l>=16 -> M=l-16, K=2/3.
// B-frag (4x16 f32): v0: K=0 (lanes 0-15, N=lane) / K=2 (lanes 16-31); v1: K=1/3.
// C/D (16x16 f32): vgpr j: M=j (lanes 0-15) / M=j+8 (lanes 16-31), N=lane%16.
// Out-of-range N columns handled via clamp + multiplicative mask so the inner
// loop has NO divergent branches (WMMA requires EXEC all-ones anyway).
// ---------------------------------------------------------------------------
__global__ void sm_iface_wmma(const float* __restrict__ xi,
                              const float* __restrict__ W,
                              const float* __restrict__ b_lin,
                              float* __restrict__ iface) {
    const int wave = threadIdx.x >> 5;
    const int lane = threadIdx.x & 31;
    const int half = lane >> 4;          // 0: lanes 0-15, 1: lanes 16-31
    const int l16  = lane & 15;

    for (int t = wave; t < 18; t += 8) {
        const int mt = t / 9;            // 0..1
        const int nt = t % 9;            // 0..8
        const int row = mt * 16 + l16;   // < 32 always
        const int n   = nt * 16 + l16;   // may exceed 137 on last tile
        const bool nok = (n < IFACE_N);
        const int  nc  = nok ? n : (IFACE_N - 1);   // clamped column
        const float msk = nok ? 1.0f : 0.0f;        // zero out OOB lanes

        const float* __restrict__ arow = xi + row * DIN + 2 * half;
        const float* __restrict__ bcol = W + nc;

        v8f c = {};
        for (int k0 = 0; k0 < DIN; k0 += 4) {
            v2f a, b;
            a.x = arow[k0];
            a.y = arow[k0 + 1];
            const int kr = k0 + 2 * half;
            b.x = msk * bcol[kr * IFACE_N];
            b.y = msk * bcol[(kr + 1) * IFACE_N];
            c = __builtin_amdgcn_wmma_f32_16x16x4_f32(
                    false, a, false, b, (short)0, c, false, false);
        }
        if (nok) {
            const float bias = b_lin[n];
            #pragma unroll
            for (int j = 0; j < 8; ++j) {
                const int mrow = mt * 16 + j + 8 * half;
                iface[mrow * IFACE_N + n] = c[j] + bias;
            }
        }
    }
}

// ---------------------------------------------------------------------------
// Kernel A2: per batch, compute ww (needs usage@rp, min, I, ig, wg) and the 9
// updated rows newrow[b,k,:] = read_vectors + ww_k * write_vector.
// 32 blocks x 64 threads.
// ---------------------------------------------------------------------------
__global__ void sm_newrows(const float* __restrict__ iface,
                           const float* __restrict__ usage,
                           const float* __restrict__ read_weights,
                           const float* __restrict__ read_vectors,
                           const int*   __restrict__ rp,
                           float* __restrict__ newrow) {
    const int b = blockIdx.x;
    const int t = threadIdx.x;
    __shared__ float s_rel[RREADS];
    __shared__ float s_ww[RREADS];
    __shared__ float s_min;

    if (t < RREADS) {
        s_rel[t] = usage[(size_t)b * MMEM + rp[b * RREADS + t]];
    }
    __syncthreads();
    if (t == 0) {
        float mn = s_rel[0];
        for (int k = 1; k < RREADS; ++k) mn = fminf(mn, s_rel[k]);
        s_min = mn;
    }
    __syncthreads();
    if (t < RREADS) {
        const float ig = 1.0f / (1.0f + expf(-iface[b * IFACE_N + 2 * WCELL + t]));
        const float wg = 1.0f / (1.0f + expf(-iface[b * IFACE_N + (IFACE_N - 1)]));
        const float I  = (s_rel[t] == s_min) ? 1.0f : 0.0f;
        const float rw = read_weights[b * RREADS + t];
        s_ww[t] = wg * (ig * rw + (1.0f - ig) * I);
    }
    __syncthreads();
    const float wv = iface[b * IFACE_N + WCELL + t];   // write_vector[t], t<64
    for (int k = 0; k < RREADS; ++k) {
        const int idx = (b * RREADS + k) * WCELL + t;
        newrow[idx] = read_vectors[idx] + s_ww[k] * wv;
    }
}

// ---------------------------------------------------------------------------
// Kernel B: full scan. s[b*M+m] = ||mem[b,m]||^2 - 2*mem[b,m].q[b]
// 4 lanes per row, float4 loads; 50000 blocks x 256 threads.
// 819 MB streamed once -> HBM-bandwidth bound (~35 us at 23.3 TB/s).
// ---------------------------------------------------------------------------
__global__ void sm_scan(const float* __restrict__ memory,
                        const float* __restrict__ iface,
                        float* __restrict__ s) {
    const size_t gtid = (size_t)blockIdx.x * 256 + threadIdx.x;
    const size_t row  = gtid >> 2;                 // global row index b*M + m
    if (row >= (size_t)BATCH * MMEM) return;
    const int c4 = (int)(gtid & 3);
    const int b  = (int)(row / MMEM);

    const float4* mrow = (const float4*)(memory + row * WCELL);
    const float* q = iface + b * IFACE_N;          // read_query = iface[:, :64]

    float acc = 0.0f;
    #pragma unroll
    for (int i = 0; i < 4; ++i) {
        const float4 x = mrow[c4 * 4 + i];
        const int cb = c4 * 16 + i * 4;
        acc += x.x * x.x - 2.0f * x.x * q[cb + 0];
        acc += x.y * x.y - 2.0f * x.y * q[cb + 1];
        acc += x.z * x.z - 2.0f * x.z * q[cb + 2];
        acc += x.w * x.w - 2.0f * x.w * q[cb + 3];
    }
    acc += __shfl_xor(acc, 1);
    acc += __shfl_xor(acc, 2);
    if (c4 == 0) s[row] = acc;
}

// ---------------------------------------------------------------------------
// Kernel C: per batch: (1) overwrite scores of the 9 updated rows (sequential
// k => duplicate rp last-wins, matching in-place set), (2) top-8 smallest with
// lowest-index tie-break via 8 exclusion arg-min passes (12.8 MB score array
// is L2-resident on re-reads), (3) gather rows with override.
// 32 blocks x 1024 threads.
// ---------------------------------------------------------------------------
__global__ void sm_topk_gather(float* __restrict__ s,
                               const float* __restrict__ memory,
                               const float* __restrict__ iface,
                               const float* __restrict__ newrow,
                               const int*   __restrict__ rp,
                               float* __restrict__ out) {
    const int b   = blockIdx.x;
    const int tid = threadIdx.x;
    __shared__ float lv[1024];
    __shared__ int   li[1024];
    __shared__ int   chosen[8];

    // Stage 0: override scores for the 9 scattered rows (wave 0 only)
    if (tid < 32) {
        const float* q = iface + b * IFACE_N;
        for (int k = 0; k < RREADS; ++k) {
            const int pos = rp[b * RREADS + k];
            const float* nr = newrow + (b * RREADS + k) * WCELL;
            float p = 0.0f;
            for (int c = tid; c < WCELL; c += 32) {
                const float x = nr[c];
                p += x * x - 2.0f * x * q[c];
            }
            p += __shfl_down(p, 16);
            p += __shfl_down(p, 8);
            p += __shfl_down(p, 4);
            p += __shfl_down(p, 2);
            p += __shfl_down(p, 1);
            if (tid == 0) s[(size_t)b * MMEM + pos] = p;
        }
    }
    __syncthreads();

    // Stage 1: 8 arg-min passes with exclusion
    const float* sb = s + (size_t)b * MMEM;
    for (int j = 0; j < 8; ++j) {
        float bv = INFINITY;
        int   bi = 0x7fffffff;
        for (int m = tid; m < MMEM; m += 1024) {
            bool skip = false;
            for (int jj = 0; jj < j; ++jj) skip |= (chosen[jj] == m);
            if (skip) continue;
            const float v = sb[m];
            if (v < bv) { bv = v; bi = m; }   // strict < keeps lowest index on ties
        }
        lv[tid] = bv; li[tid] = bi;
        __syncthreads();
        for (int off = 512; off > 0; off >>= 1) {
            if (tid < off) {
                const float v2 = lv[tid + off];
                const int   i2 = li[tid + off];
                if (v2 < lv[tid] || (v2 == lv[tid] && i2 < li[tid])) {
                    lv[tid] = v2; li[tid] = i2;
                }
            }
            __syncthreads();
        }
        if (tid == 0) chosen[j] = li[0];
        __syncthreads();
    }

    // Stage 2: gather rows (with rp-override, last k wins)
    for (int idx = tid; idx < 8 * WCELL; idx += 1024) {
        const int j = idx >> 6;
        const int c = idx & 63;
        const int pos = chosen[j];
        int kk = -1;
        for (int k = 0; k < RREADS; ++k)
            if (rp[b * RREADS + k] == pos) kk = k;
        float val;
        if (kk >= 0) val = newrow[(b * RREADS + kk) * WCELL + c];
        else         val = memory[((size_t)b * MMEM + pos) * WCELL + c];
        out[(b * 8 + j) * WCELL + c] = val;
    }
}

// ---------------------------------------------------------------------------
extern "C" void kernel_launch(void* const* d_in, const int* in_sizes, int n_in,
                              void* d_out, int out_size, void* d_ws, size_t ws_size,
                              hipStream_t stream) {
    const float* xi           = (const float*)d_in[0];
    const float* W            = (const float*)d_in[1];
    const float* b_lin        = (const float*)d_in[2];
    const float* memory       = (const float*)d_in[3];
    const float* usage        = (const float*)d_in[4];
    const float* read_weights = (const float*)d_in[5];
    // d_in[6] write_weights : dead (only feeds discarded usage update)
    const float* read_vectors = (const float*)d_in[7];
    // d_in[8] last_used_mem : dead (output drops last row)
    const int*   read_pos     = (const int*)d_in[9];
    // d_in[10] timestep     : dead
    (void)in_sizes; (void)n_in; (void)out_size; (void)ws_size;

    float* wsf    = (float*)d_ws;
    float* iface  = wsf + WS_IFACE;
    float* newrow = wsf + WS_NEWROW;
    float* scores = wsf + WS_SCORES;
    float* out    = (float*)d_out;

    sm_iface_wmma<<<1, 256, 0, stream>>>(xi, W, b_lin, iface);
    sm_newrows<<<BATCH, 64, 0, stream>>>(iface, usage, read_weights,
                                         read_vectors, read_pos, newrow);
    // rows*4 lanes = 32*100000*4 = 12,800,000 threads -> exactly 50000 blocks
    sm_scan<<<50000, 256, 0, stream>>>(memory, iface, scores);
    sm_topk_gather<<<BATCH, 1024, 0, stream>>>(scores, memory, iface,
                                               newrow, read_pos, out);
}